// FocalLoss_9612136808648
// MI455X (gfx1250) — compile-verified
//
#include <hip/hip_runtime.h>
#include <hip/hip_bf16.h>

typedef __attribute__((ext_vector_type(2))) float v2f;
typedef __attribute__((ext_vector_type(8))) float v8f;

// Problem constants (from the reference setup)
#define NB    8
#define NM    128
#define NC    8
#define NA    47616              // 3*(8192+4096+2048+1024+512)
#define NW    1488               // NA/32 bitmask words per sample
#define NCHUNK 16
#define APC   (NA / NCHUNK)      // 2976 anchors per focal block

// ---------------------------------------------------------------------------
// Wave-wide f32 reduction on the CDNA5 matrix pipe:
// A(16x4) holds the 32 per-lane partials (va = {x, 0} -> 2 VGPRs/lane),
// B(4x16) = ones, D(16x16) rows = row-sums. Sum the 8 D VGPRs per lane,
// lanes 0-15 hold rows 0-7, lanes 16-31 rows 8-15; one shfl_xor(16) finishes.
// EXEC must be all-ones at the call site (full block, converged).
// ---------------------------------------------------------------------------
__device__ __forceinline__ float wave_reduce_wmma(float x) {
    v2f a;  a[0] = x;    a[1] = 0.0f;
    v2f bo; bo[0] = 1.0f; bo[1] = 1.0f;
    v8f c = {};
    c = __builtin_amdgcn_wmma_f32_16x16x4_f32(false, a, false, bo,
                                              (short)0, c, false, false);
    float t = c[0] + c[1] + c[2] + c[3] + c[4] + c[5] + c[6] + c[7];
    t += __shfl_xor(t, 16, 32);
    return t;   // every lane holds the full 32-lane sum
}

// ---------------------------------------------------------------------------
// Kernel 0: zero workspace (mask + npos + S)
// ---------------------------------------------------------------------------
__global__ void zero_ws_kernel(unsigned int* __restrict__ ws, int n) {
    int i = blockIdx.x * blockDim.x + threadIdx.x;
    if (i < n) ws[i] = 0u;
}

// ---------------------------------------------------------------------------
// Kernel 1: ATSS assignment. One thread per (b, gt).
// Exploits uniform anchor grid: top-27-by-distance per level == the 9 nearest
// locations x 3 scales; 9 nearest of sorted uniform centers = contiguous
// window found by greedy expansion (tie -> left, matching top_k index order).
// Three passes over the 135 candidates: sum -> ssd (two-pass std) -> test.
// Positives OR'd into a per-sample anchor bitmask.
// ---------------------------------------------------------------------------
__global__ __launch_bounds__(NM) void atss_assign_kernel(
        const float* __restrict__ ann, unsigned int* __restrict__ mask) {
    const int b = blockIdx.x;
    const int m = threadIdx.x;
    const float* g = ann + ((size_t)b * NM + m) * 3;
    const float g0 = g[0], g1 = g[1];
    const float gcx = (g0 + g1) * 0.5f;
    const float gw  = g1 - g0;

    const int   Ls[5] = {8192, 4096, 2048, 1024, 512};
    const float ss[5] = {8.f, 16.f, 32.f, 64.f, 128.f};
    const int   bs[5] = {0, 24576, 36864, 43008, 46080};

    int lo5[5];
#pragma unroll
    for (int l = 0; l < 5; ++l) {
        const float s = ss[l];
        const int   L = Ls[l];
        float x = gcx / s - 0.5f;
        int i0 = (int)floorf(x + 0.5f);
        i0 = i0 < 0 ? 0 : (i0 > L - 1 ? L - 1 : i0);
        int lo = i0, hi = i0;
#pragma unroll
        for (int t = 0; t < 8; ++t) {
            float dl = (lo > 0)     ? fabsf(((float)lo - 0.5f) * s - gcx) : 3.4e38f;
            float dr = (hi < L - 1) ? fabsf(((float)hi + 1.5f) * s - gcx) : 3.4e38f;
            if (dl <= dr) --lo; else ++hi;
        }
        lo5[l] = lo;
    }

    // Pass A: mean of 135 candidate IoUs
    float sum = 0.0f;
#pragma unroll
    for (int l = 0; l < 5; ++l) {
        const float s = ss[l];
        for (int j = 0; j < 9; ++j) {
            const float c = ((float)(lo5[l] + j) + 0.5f) * s;
#pragma unroll
            for (int sc = 0; sc < 3; ++sc) {
                const float wh = (2.0f + (float)sc) * s;   // half-width: (4+2sc)/2 * s
                const float a0 = c - wh, a1 = c + wh;
                float iw = fmaxf(fminf(a1, g1) - fmaxf(a0, g0), 0.0f);
                float ua = fmaxf((a1 - a0) + gw - iw, 1e-8f);
                sum += iw / ua;
            }
        }
    }
    const float mean = sum / 135.0f;

    // Pass B: unbiased std (two-pass, good numerics)
    float ssd = 0.0f;
#pragma unroll
    for (int l = 0; l < 5; ++l) {
        const float s = ss[l];
        for (int j = 0; j < 9; ++j) {
            const float c = ((float)(lo5[l] + j) + 0.5f) * s;
#pragma unroll
            for (int sc = 0; sc < 3; ++sc) {
                const float wh = (2.0f + (float)sc) * s;
                const float a0 = c - wh, a1 = c + wh;
                float iw = fmaxf(fminf(a1, g1) - fmaxf(a0, g0), 0.0f);
                float ua = fmaxf((a1 - a0) + gw - iw, 1e-8f);
                float d = iw / ua - mean;
                ssd += d * d;
            }
        }
    }
    const float thresh = mean + sqrtf(ssd / 134.0f);

    // Pass C: threshold + center-inside test, scatter to bitmask
#pragma unroll
    for (int l = 0; l < 5; ++l) {
        const float s = ss[l];
        for (int j = 0; j < 9; ++j) {
            const int   loc = lo5[l] + j;
            const float c   = ((float)loc + 0.5f) * s;
#pragma unroll
            for (int sc = 0; sc < 3; ++sc) {
                const float wh = (2.0f + (float)sc) * s;
                const float a0 = c - wh, a1 = c + wh;
                float iw = fmaxf(fminf(a1, g1) - fmaxf(a0, g0), 0.0f);
                float ua = fmaxf((a1 - a0) + gw - iw, 1e-8f);
                float iou = iw / ua;
                if (iou >= thresh && fminf(c - g0, g1 - c) > 0.01f) {
                    int idx = bs[l] + loc * 3 + sc;
                    atomicOr(&mask[b * NW + (idx >> 5)], 1u << (idx & 31));
                }
            }
        }
    }
}

// ---------------------------------------------------------------------------
// Kernel 2: n_pos per sample via popcount of the bitmask
// ---------------------------------------------------------------------------
__global__ void npos_kernel(const unsigned int* __restrict__ mask,
                            int* __restrict__ npos) {
    int i = blockIdx.x * blockDim.x + threadIdx.x;
    if (i < NB * NW) {
        unsigned int w = mask[i];
        if (w) atomicAdd(&npos[i / NW], __popc(w));
    }
}

// ---------------------------------------------------------------------------
// Kernel 3: focal-loss partial sums. grid = (NCHUNK, NB), 256 threads.
// Per anchor: one bitmask bit + two float4 loads (8 classes, 32B aligned).
// Block reduction rides the WMMA f32 16x16x4 matrix pipe.
// ---------------------------------------------------------------------------
__global__ __launch_bounds__(256) void focal_kernel(
        const float* __restrict__ cls, const unsigned int* __restrict__ mask,
        const int* __restrict__ class_id_p, float* __restrict__ S) {
    const int b   = blockIdx.y;
    const int a0  = blockIdx.x * APC;
    const int a1e = a0 + APC;
    const int cid = class_id_p[0];
    const float4* base = (const float4*)(cls + (size_t)b * NA * NC);

    float acc = 0.0f;
    for (int a = a0 + threadIdx.x; a < a1e; a += blockDim.x) {
        const unsigned int w = mask[b * NW + (a >> 5)];
        const bool pos = (w >> (a & 31)) & 1u;
        float4 p0 = base[a * 2 + 0];
        float4 p1 = base[a * 2 + 1];
        float pv[8] = {p0.x, p0.y, p0.z, p0.w, p1.x, p1.y, p1.z, p1.w};
#pragma unroll
        for (int c = 0; c < 8; ++c) {
            float p = fminf(fmaxf(pv[c], 1e-4f), 1.0f - 1e-4f);
            if (pos && c == cid)
                acc += 0.25f * (1.0f - p) * (1.0f - p) * (-logf(p));
            else
                acc += 0.75f * p * p * (-logf(1.0f - p));
        }
    }

    // all 256 threads converged here -> EXEC all-ones for WMMA
    float t = wave_reduce_wmma(acc);

    __shared__ float wsum[8];
    const int lane = threadIdx.x & 31;
    const int wid  = threadIdx.x >> 5;
    if (lane == 0) wsum[wid] = t;
    __syncthreads();
    if (threadIdx.x == 0) {
        float tot = 0.0f;
#pragma unroll
        for (int i = 0; i < 8; ++i) tot += wsum[i];
        atomicAdd(&S[b], tot);
    }
}

// ---------------------------------------------------------------------------
// Kernel 4: finalize scalar
// ---------------------------------------------------------------------------
__global__ void final_kernel(const float* __restrict__ S,
                             const int* __restrict__ npos,
                             float* __restrict__ out) {
    if (threadIdx.x == 0 && blockIdx.x == 0) {
        float acc = 0.0f;
#pragma unroll
        for (int b = 0; b < NB; ++b)
            acc += S[b] / fmaxf((float)npos[b], 1.0f);
        out[0] = acc * (1.0f / (float)NB);
    }
}

extern "C" void kernel_launch(void* const* d_in, const int* in_sizes, int n_in,
                              void* d_out, int out_size, void* d_ws, size_t ws_size,
                              hipStream_t stream) {
    const float* cls = (const float*)d_in[0];   // [B, A, C] f32
    const float* ann = (const float*)d_in[1];   // [B, M, 3] f32
    const int*   cid = (const int*)d_in[7];     // scalar class_id

    unsigned int* mask = (unsigned int*)d_ws;            // NB*NW words
    int*          npos = (int*)(mask + NB * NW);         // NB ints
    float*        S    = (float*)(npos + NB);            // NB floats
    float*        out  = (float*)d_out;

    const int nzero = NB * NW + NB + NB;
    zero_ws_kernel<<<(nzero + 255) / 256, 256, 0, stream>>>((unsigned int*)d_ws, nzero);
    atss_assign_kernel<<<NB, NM, 0, stream>>>(ann, mask);
    npos_kernel<<<(NB * NW + 255) / 256, 256, 0, stream>>>(mask, npos);
    focal_kernel<<<dim3(NCHUNK, NB), 256, 0, stream>>>(cls, mask, cid, S);
    final_kernel<<<1, 32, 0, stream>>>(S, npos, out);
}